// RecallatK_13142599926012
// MI455X (gfx1250) — compile-verified
//
#include <hip/hip_runtime.h>
#include <hip/hip_bf16.h>
#include <cstdint>

#define BDIM 2048
#define DDIM 1024
#define MQ   16          // queries per block (2 groups of 8)
#define NW   8           // waves per block
#define CHUNKF 258       // 256 data floats + 2 pad floats (matches TDM pad_interval/amount)
#define ROWF   (4 * CHUNKF)   // 1032 floats per padded row; 1032 % 64 == 8 -> 2-way banks

typedef float v2f __attribute__((ext_vector_type(2)));
typedef float v8f __attribute__((ext_vector_type(8)));
typedef unsigned int u32x4 __attribute__((ext_vector_type(4)));
typedef int i32x4 __attribute__((ext_vector_type(4)));
typedef int i32x8 __attribute__((ext_vector_type(8)));

#if __has_builtin(__builtin_amdgcn_tensor_load_to_lds)
#define USE_TDM 1
#warning "CDNA5 probe: TDM builtin AVAILABLE - tensor_load_to_lds path compiled"
#else
#define USE_TDM 0
#warning "CDNA5 probe: TDM builtin NOT available - manual LDS staging fallback"
#endif

__device__ __forceinline__ float fast_sig(float e) {
    // 1/(1+exp(e)) with e clamped to [-50,50]
    e = fminf(fmaxf(e, -50.0f), 50.0f);
    return __builtin_amdgcn_rcpf(1.0f + __builtin_amdgcn_exp2f(e * 1.44269504088896340f));
}

__global__ __launch_bounds__(256) void recall_rk_kernel(const float* __restrict__ preds,
                                                        float* __restrict__ partials) {
    extern __shared__ float lds[];
    float* qs     = lds;                    // MQ * ROWF        padded query rows
    float* gbuf   = qs + MQ * ROWF;         // MQ * 8           G matrix (later reused as rk)
    float* rkpart = gbuf + MQ * 8;          // NW * MQ * 8      per-wave rk partials
    float* qloss  = rkpart + NW * MQ * 8;   // MQ

    const int tid  = threadIdx.x;
    const int q0   = blockIdx.x * MQ;
    const int lane = tid & 31;
    const int wv   = tid >> 5;
    const int h    = lane >> 4;      // K-slot-pair selector for WMMA f32 frags
    const int m    = lane & 15;      // row (A) / col (B) index within tile

    // ---- stage 16 query rows into padded LDS layout ----
#if USE_TDM
    if (wv == 0) {
        const uint64_t ga = (uint64_t)(uintptr_t)(preds + (size_t)q0 * DDIM);
        const uint32_t la = (uint32_t)(uintptr_t)qs;   // LDS byte offset (low 32 bits)
        u32x4 g0;
        g0[0] = 1u;                                    // count=1, is_restore=0
        g0[1] = la;                                    // lds_addr (bytes)
        g0[2] = (uint32_t)(ga & 0xFFFFFFFFu);          // global_addr[31:0]
        g0[3] = (uint32_t)((ga >> 32) & 0x01FFFFFFu) | (2u << 30);  // addr[56:32] | type=2
        i32x8 g1;
        g1[0] = (int)((2u << 16)    // data_size = 4B
                    | (1u << 20)    // pad_enable
                    | (7u << 22)    // pad_interval: 256 DWORDs
                    | (1u << 25));  // pad_amount: 2 DWORDs
        g1[1] = (int)(1024u << 16);  // tensor_dim0 = 1024 (bits 79:48)
        g1[2] = (int)(2048u << 16);  // tensor_dim1 = 2048 (bits 111:80)
        g1[3] = (int)(1024u << 16);  // tile_dim0 = 1024 (bits 127:112)
        g1[4] = 16;                  // tile_dim1 = 16
        g1[5] = 1024;                // tensor_dim0_stride = 1024
        g1[6] = 0;
        g1[7] = 0;
        i32x4 gz = {0, 0, 0, 0};
#if defined(__clang_major__) && (__clang_major__ >= 23)
        i32x8 gz8 = {0, 0, 0, 0, 0, 0, 0, 0};
        __builtin_amdgcn_tensor_load_to_lds(g0, g1, gz, gz, gz8, 0);
#else
        __builtin_amdgcn_tensor_load_to_lds(g0, g1, gz, gz, 0);
#endif
        __builtin_amdgcn_s_wait_tensorcnt(0);
    }
#else
    for (int idx = tid; idx < MQ * (DDIM / 2); idx += 256) {
        int row = idx >> 9;                 // DDIM/2 == 512
        int c2  = (idx & 511) * 2;
        v2f v = *(const v2f*)(preds + (size_t)(q0 + row) * DDIM + c2);
        *(v2f*)(qs + row * ROWF + (c2 >> 8) * CHUNKF + (c2 & 255)) = v;
    }
#endif
    __syncthreads();

    // ---- wave 0: diagonal 16x16 tile -> G[q][i] (A-frag == B-frag) ----
    if (wv == 0) {
        v8f accd = {};
        #pragma unroll
        for (int seg = 0; seg < 4; ++seg) {
            const float* aseg = qs + m * ROWF + seg * CHUNKF + 4 * h;
            #pragma unroll 4
            for (int kk = 0; kk < 256; kk += 8) {
                v2f a01 = *(const v2f*)(aseg + kk);
                v2f a23 = *(const v2f*)(aseg + kk + 2);
                accd = __builtin_amdgcn_wmma_f32_16x16x4_f32(false, a01, false, a01,
                                                             (short)0, accd, false, false);
                accd = __builtin_amdgcn_wmma_f32_16x16x4_f32(false, a23, false, a23,
                                                             (short)0, accd, false, false);
            }
        }
        // C layout: VGPR r -> M = r + 8h, N = lane&15. G cols: 0..7 for M<8, else 8..15.
        int i = m - 8 * h;
        if (i >= 0 && i < 8) {
            #pragma unroll
            for (int r = 0; r < 8; ++r) gbuf[(r + 8 * h) * 8 + i] = accd[r];
        }
    }
    __syncthreads();

    // ---- preload G rows this lane needs (q_local = r + 8h) ----
    float gv[64];
    #pragma unroll
    for (int r = 0; r < 8; ++r)
        #pragma unroll
        for (int i = 0; i < 8; ++i)
            gv[r * 8 + i] = gbuf[(r + 8 * h) * 8 + i];

    float rk[64];
    #pragma unroll
    for (int x = 0; x < 64; ++x) rk[x] = 0.0f;

    // ---- stream column tiles: b128 loads feed 2 WMMAs (K=8), fuse sigmoid-sum ----
    for (int t = wv; t < BDIM / 16; t += NW) {
        const int c0 = t * 16;
        const float* bp = preds + (size_t)(c0 + m) * DDIM + 4 * h;
        if (t + NW < BDIM / 16)   // prefetch next tile's row for this lane
            __builtin_prefetch(preds + (size_t)(c0 + NW * 16 + m) * DDIM, 0, 3);
        v8f acc = {};
        #pragma unroll
        for (int seg = 0; seg < 4; ++seg) {
            const float* aseg = qs + m * ROWF + seg * CHUNKF + 4 * h;
            const float* bseg = bp + seg * 256;
            #pragma unroll 4
            for (int kk = 0; kk < 256; kk += 8) {
                float4 b4 = *(const float4*)(bseg + kk);     // global_load_b128
                v2f a01 = *(const v2f*)(aseg + kk);          // ds_load_2addr_b64 pair
                v2f a23 = *(const v2f*)(aseg + kk + 2);
                v2f b01 = {b4.x, b4.y};
                v2f b23 = {b4.z, b4.w};
                // K-permuted accumulation: slots {01}= {k,k+1}|{k+4,k+5}, {23}= rest
                acc = __builtin_amdgcn_wmma_f32_16x16x4_f32(false, a01, false, b01,
                                                            (short)0, acc, false, false);
                acc = __builtin_amdgcn_wmma_f32_16x16x4_f32(false, a23, false, b23,
                                                            (short)0, acc, false, false);
            }
        }
        const int j = c0 + m;              // this lane's column
        const int d = j - (q0 + 8 * h);    // diag-mask hits when d == i (0..7)
        #pragma unroll
        for (int r = 0; r < 8; ++r) {
            const float s = acc[r];        // S[q0 + r + 8h][j]
            #pragma unroll
            for (int i = 0; i < 8; ++i) {
                // sigmoid((S - G)/0.01) = 1/(1+exp(clamp((G - S)*100)))
                float sg = fast_sig((gv[r * 8 + i] - s) * 100.0f);
                rk[r * 8 + i] += (d == i) ? 0.0f : sg;
            }
        }
    }

    // ---- reduce rk across the 16 lanes of each half-wave ----
    #pragma unroll
    for (int x = 0; x < 64; ++x) {
        float v = rk[x];
        v += __shfl_xor(v, 1, 32);
        v += __shfl_xor(v, 2, 32);
        v += __shfl_xor(v, 4, 32);
        v += __shfl_xor(v, 8, 32);
        rk[x] = v;
    }
    if (m == 0) {  // lanes 0 (queries q0..q0+7) and 16 (queries q0+8..q0+15)
        #pragma unroll
        for (int x = 0; x < 64; ++x) {
            int r = x >> 3, i = x & 7;
            rkpart[(wv * MQ + 8 * h + r) * 8 + i] = rk[x];
        }
    }
    __syncthreads();

    // ---- combine waves: rk[q][i] = 1 + sum_w partial (reuse gbuf) ----
    if (tid < MQ * 8) {
        int q = tid >> 3, i = tid & 7;
        float s = 0.0f;
        #pragma unroll
        for (int w = 0; w < NW; ++w) s += rkpart[(w * MQ + q) * 8 + i];
        gbuf[q * 8 + i] = 1.0f + s;
    }
    __syncthreads();

    // ---- per-query recall@k tail ----
    if (tid < MQ) {
        const float kvv[5] = {1.f, 2.f, 4.f, 8.f, 16.f};
        const float ktv[5] = {1.f, 2.f, 4.f, 8.f, 16.f};
        const float nvv[5] = {1.f, 2.f, 4.f, 7.f, 7.f};   // min(k, spc-1)
        const int rself = tid & 7;
        float vsum[5] = {0.f, 0.f, 0.f, 0.f, 0.f};
        for (int i = 0; i < 8; ++i) {
            if (i == rself) continue;                      // self mask
            float rkv = gbuf[tid * 8 + i];
            #pragma unroll
            for (int k = 0; k < 5; ++k)
                vsum[k] += fast_sig((rkv - kvv[k]) / ktv[k]);
        }
        float rec = 0.0f;
        #pragma unroll
        for (int k = 0; k < 5; ++k) rec += fminf(vsum[k], kvv[k]) / nvv[k];
        qloss[tid] = 1.0f - rec * 0.2f;
    }
    __syncthreads();
    if (tid == 0) {
        float s = 0.0f;
        for (int q = 0; q < MQ; ++q) s += qloss[q];
        partials[blockIdx.x] = s;
    }
}

__global__ __launch_bounds__(128) void recall_reduce_kernel(const float* __restrict__ partials,
                                                            float* __restrict__ out) {
    __shared__ float sm[128];
    const int tid = threadIdx.x;
    sm[tid] = partials[tid];
    __syncthreads();
    for (int off = 64; off > 0; off >>= 1) {
        if (tid < off) sm[tid] += sm[tid + off];
        __syncthreads();
    }
    if (tid == 0) out[0] = sm[0] * (1.0f / (float)BDIM);
}

extern "C" void kernel_launch(void* const* d_in, const int* in_sizes, int n_in,
                              void* d_out, int out_size, void* d_ws, size_t ws_size,
                              hipStream_t stream) {
    const float* preds = (const float*)d_in[0];   // [2048,1024] f32 (q_idx unused)
    float* partials = (float*)d_ws;               // 128 floats of scratch
    float* out = (float*)d_out;                   // scalar f32 loss

    const size_t shmem = (size_t)(MQ * ROWF + MQ * 8 + NW * MQ * 8 + MQ) * sizeof(float);
    hipLaunchKernelGGL(recall_rk_kernel, dim3(BDIM / MQ), dim3(256), shmem, stream,
                       preds, partials);
    hipLaunchKernelGGL(recall_reduce_kernel, dim3(1), dim3(128), 0, stream,
                       partials, out);
}